// MultiScaleEncoder_59047210385598
// MI455X (gfx1250) — compile-verified
//
#include <hip/hip_runtime.h>
#include <cstdint>
#include <cstddef>

// ---------------------------------------------------------------------------
// MultiScaleEncoder for MI455X (gfx1250).
// Dense linears -> V_WMMA_F32_16X16X4_F32 (fp32 WMMA; workload is HBM-bound at
// 23.3 TB/s, AI ~32 flop/B, so fp32 matrix pipes are free - no need for bf16).
// GEMM v3: 256-thr block = 8 waves, each wave owns a 16-row strip and ALL
// output columns (4/8 accumulator tiles) -> 8x A reuse; weights staged
// pair-packed in LDS; B fragments kept simultaneously live so DS loads clause
// and wait once per k-step; A fragment software-pipelined one step ahead.
// ---------------------------------------------------------------------------

#define N_NODES 100000
#define N_EDGES 400000
#define N_GRAPH 2048
#define F_DIM   128
#define H_DIM   64
#define MAXN_T  128
#define HEADS_T 8

typedef float v2f __attribute__((ext_vector_type(2)));
typedef float v8f __attribute__((ext_vector_type(8)));

// ------------------------------ WMMA GEMM ----------------------------------
// Y[M, NT*16] = act( X[M,din] @ W[din, NT*16] + bias ), row-major.
// grid.x = ceil(M/128); wave w handles rows [128*bx + 16w, +16), all columns.
// A layout (16x4 f32): lane<16 -> row=m0+lane, k={k0,k0+1}; lane>=16 -> k={k0+2,k0+3}.
// B fragment symmetric; C/D: vgpr i -> row m0+half*8+i, col n0+(lane&15).
template <int NT>
__global__ __launch_bounds__(256)
void gemm_v3(const float* __restrict__ X, const float* __restrict__ W,
             const float* __restrict__ bias, float* __restrict__ Y,
             int M, int din, int act)
{
    const int dout = NT * 16;
    extern __shared__ float Ws[];            // pair-packed: Kc/2 * dout * 2 floats
    const int tid  = threadIdx.x;
    const int lane = tid & 31;
    const int wave = tid >> 5;
    const int half = lane >> 4;
    const int l15  = lane & 15;
    const int m0   = blockIdx.x * 128 + wave * 16;
    int rowA = m0 + l15; if (rowA >= M) rowA = M - 1;      // per-lane select, EXEC intact
    const float* __restrict__ Xr = X + (size_t)rowA * din;
    const int Kc = din < 128 ? din : 128;

    v8f acc[NT];
    const v8f zero = {0.f, 0.f, 0.f, 0.f, 0.f, 0.f, 0.f, 0.f};
#pragma unroll
    for (int t = 0; t < NT; ++t) acc[t] = zero;

    for (int kc = 0; kc < din; kc += Kc) {
        // stage W[kc..kc+Kc) into LDS, pair-packed along K:
        // Ws[(kk*dout + c)*2 + {0,1}] = W[kc+2kk][c], W[kc+2kk+1][c]
        for (int idx = tid; idx < (Kc / 2) * dout; idx += 256) {
            int kk = idx / dout, c = idx - kk * dout;
            const float* wp = W + ((size_t)(kc + 2 * kk) * dout + c);
            Ws[idx * 2]     = wp[0];
            Ws[idx * 2 + 1] = wp[dout];
        }
        if (kc + Kc < din)                    // CDNA5 prefetch of next W chunk
            __builtin_prefetch(W + (size_t)(kc + Kc) * dout + tid * 16, 0, 1);
        __syncthreads();

        const int iters = Kc / 4;
        // software-pipelined A fragment (global) + clustered B fragments (LDS)
        v2f a_next = *(const v2f*)&Xr[kc + half * 2];
#pragma unroll 2
        for (int i = 0; i < iters; ++i) {
            const v2f a = a_next;
            if (i + 1 < iters)                // issue next A load before the MMAs
                a_next = *(const v2f*)&Xr[kc + 4 * (i + 1) + half * 2];
            const int brow = (2 * i + half) * dout;
            v2f bfrag[NT];
#pragma unroll
            for (int t = 0; t < NT; ++t)      // all NT ds_loads issued, one wait
                bfrag[t] = *(const v2f*)&Ws[(brow + t * 16 + l15) * 2];
#pragma unroll
            for (int t = 0; t < NT; ++t)
                acc[t] = __builtin_amdgcn_wmma_f32_16x16x4_f32(
                             false, a, false, bfrag[t], (short)0, acc[t], false, false);
        }
        __syncthreads();
    }

    const int rowbase = m0 + half * 8;
#pragma unroll
    for (int t = 0; t < NT; ++t) {
        const int col = t * 16 + l15;
        const float bn = bias ? bias[col] : 0.f;
#pragma unroll
        for (int i = 0; i < 8; ++i) {
            const int r = rowbase + i;
            if (r < M) {
                float v = acc[t][i] + bn;
                if (act == 1)      v = fmaxf(v, 0.f);
                else if (act == 2) v = 1.f / (1.f + __expf(-v));
                Y[(size_t)r * dout + col] = v;
            }
        }
    }
}

// --------------------------- elementwise kernels ---------------------------
#define TGRID int t = blockIdx.x * blockDim.x + threadIdx.x;

__global__ void embed_kernel(const int* __restrict__ xa, const int* __restrict__ xc,
                             const float* __restrict__ ea, const float* __restrict__ ec,
                             float* __restrict__ out, int n) {
    TGRID if (t >= n) return;
    int i = t / H_DIM, f = t % H_DIM;
    out[t] = ea[xa[i] * H_DIM + f] + ec[xc[i] * H_DIM + f];
}

__global__ void deg_kernel(const int* __restrict__ dst, float* deg, int e) {
    TGRID if (t >= e) return;
    atomicAdd(&deg[dst[t]], 1.f);
}
__global__ void dis_kernel(float* deg, int n) {            // in place: rsqrt(deg+1)
    TGRID if (t >= n) return;
    deg[t] = rsqrtf(deg[t] + 1.f);
}
__global__ void coef_kernel(const int* __restrict__ src, const int* __restrict__ dst,
                            const float* __restrict__ dis, float* coef, int e) {
    TGRID if (t >= e) return;
    coef[t] = dis[src[t]] * dis[dst[t]];
}
__global__ void scatter_edges(const float* __restrict__ h, const int* __restrict__ src,
                              const int* __restrict__ dst, const float* __restrict__ coef,
                              float* agg, int total) {
    TGRID if (t >= total) return;
    int e = t / F_DIM, f = t % F_DIM;
    atomicAdd(&agg[(size_t)dst[e] * F_DIM + f], h[(size_t)src[e] * F_DIM + f] * coef[e]);
}
__global__ void gcn_finish(const float* __restrict__ agg, const float* __restrict__ h,
                           const float* __restrict__ dis, const float* __restrict__ b,
                           float* out, int n) {
    TGRID if (t >= n) return;
    int i = t / F_DIM, f = t % F_DIM;
    float d = dis[i];
    out[t] = agg[t] + h[t] * d * d + b[f];
}
__global__ void cnt_kernel(const int* __restrict__ batch, float* cnt, int n) {
    TGRID if (t >= n) return;
    atomicAdd(&cnt[batch[t]], 1.f);
}
__global__ void seg_sum_feat(const float* __restrict__ x, const int* __restrict__ batch,
                             float* acc, int n) {
    TGRID if (t >= n) return;
    int i = t / F_DIM, f = t % F_DIM;
    atomicAdd(&acc[(size_t)batch[i] * F_DIM + f], x[t]);
}
__global__ void gn_center(const float* __restrict__ x, const int* __restrict__ batch,
                          const float* __restrict__ acc, const float* __restrict__ cnt,
                          const float* __restrict__ ms, float* out, float* vacc, int n) {
    TGRID if (t >= n) return;
    int i = t / F_DIM, f = t % F_DIM, g = batch[i];
    float c = fmaxf(cnt[g], 1.f);
    float o = x[t] - (acc[(size_t)g * F_DIM + f] / c) * ms[f];
    out[t] = o;
    atomicAdd(&vacc[(size_t)g * F_DIM + f], o * o);
}
__global__ void gn_final(const float* __restrict__ xc, const int* __restrict__ batch,
                         const float* __restrict__ vacc, const float* __restrict__ cnt,
                         const float* __restrict__ w, const float* __restrict__ b,
                         float* y, int n) {
    TGRID if (t >= n) return;
    int i = t / F_DIM, f = t % F_DIM, g = batch[i];
    float c = fmaxf(cnt[g], 1.f);
    y[t] = xc[t] * rsqrtf(vacc[(size_t)g * F_DIM + f] / c + 1e-5f) * w[f] + b[f];
}
__global__ void pn_center(const float* __restrict__ x, const int* __restrict__ batch,
                          const float* __restrict__ acc, const float* __restrict__ cnt,
                          float* out, float* msq, int n) {
    TGRID if (t >= n) return;
    int i = t / F_DIM, f = t % F_DIM, g = batch[i];
    float c = fmaxf(cnt[g], 1.f);
    float o = x[t] - acc[(size_t)g * F_DIM + f] / c;
    out[t] = o;
    atomicAdd(&msq[g], o * o);
}
__global__ void pn_final_relu(const float* __restrict__ xc, const int* __restrict__ batch,
                              const float* __restrict__ msq, const float* __restrict__ cnt,
                              float* y, int n) {
    TGRID if (t >= n) return;
    int i = t / F_DIM, g = batch[i];
    float c = fmaxf(cnt[g], 1.f);
    y[t] = fmaxf(xc[t] * rsqrtf(1e-5f + msq[g] / c), 0.f);
}
__global__ void bn_stats(const float* __restrict__ X, float* acc, int R, int C) {
    int f = blockIdx.x;        // one block per feature
    int t = threadIdx.x;
    float s = 0.f, s2 = 0.f;
    for (int r = t; r < R; r += blockDim.x) {
        float v = X[(size_t)r * C + f];
        s += v; s2 += v * v;
    }
    __shared__ float sh[256], sh2[256];
    sh[t] = s; sh2[t] = s2; __syncthreads();
    for (int st = 128; st > 0; st >>= 1) {
        if (t < st) { sh[t] += sh[t + st]; sh2[t] += sh2[t + st]; }
        __syncthreads();
    }
    if (t == 0) { acc[f] = sh[0]; acc[C + f] = sh2[0]; }
}
__global__ void bn_apply(const float* __restrict__ X, const float* __restrict__ acc,
                         const float* __restrict__ g, const float* __restrict__ b,
                         float* Y, int R, int C) {
    TGRID if (t >= R * C) return;
    int f = t % C;
    float m = acc[f] / (float)R;
    float v = acc[C + f] / (float)R - m * m;
    Y[t] = (X[t] - m) * rsqrtf(v + 1e-5f) * g[f] + b[f];
}
__global__ void add_ln_act(const float* __restrict__ A, const float* __restrict__ B,
                           const float* __restrict__ g, const float* __restrict__ b,
                           float* __restrict__ Y, int act) {    // blockDim == F_DIM
    int r = blockIdx.x, f = threadIdx.x;
    float v = A[(size_t)r * F_DIM + f] + B[(size_t)r * F_DIM + f];
    __shared__ float sh[F_DIM];
    sh[f] = v; __syncthreads();
    for (int s = F_DIM / 2; s > 0; s >>= 1) { if (f < s) sh[f] += sh[f + s]; __syncthreads(); }
    float m = sh[0] / (float)F_DIM; __syncthreads();
    float d = v - m;
    sh[f] = d * d; __syncthreads();
    for (int s = F_DIM / 2; s > 0; s >>= 1) { if (f < s) sh[f] += sh[f + s]; __syncthreads(); }
    float var = sh[0] / (float)F_DIM;
    float y = d * rsqrtf(var + 1e-5f) * g[f] + b[f];
    if (act) y = fmaxf(y, 0.f);
    Y[(size_t)r * F_DIM + f] = y;
}
__global__ void ew_sigmul(const float* __restrict__ q, const float* __restrict__ k,
                          const float* __restrict__ v, float* o, int n) {
    TGRID if (t >= n) return;
    o[t] = v[t] / (1.f + __expf(-(q[t] * k[t])));
}
__global__ void concat3(const float* __restrict__ a, const float* __restrict__ b,
                        const float* __restrict__ c, float* cc, int n) {   // n = N*3F
    TGRID if (t >= n) return;
    int i = t / (3 * F_DIM), j = t % (3 * F_DIM);
    const float* s = (j < F_DIM) ? a : ((j < 2 * F_DIM) ? b : c);
    cc[t] = s[(size_t)i * F_DIM + (j & (F_DIM - 1))];
}
__global__ void fuse_gate(const float* __restrict__ gate, const float* __restrict__ fin,
                          const float* __restrict__ xm, float* out, int n) {
    TGRID if (t >= n) return;
    float gg = gate[t];
    out[t] = gg * fin[t] + (1.f - gg) * xm[t];
}
__global__ void starts_kernel(const int* __restrict__ batch, int* starts, int n) {
    TGRID if (t >= n) return;
    if (t == 0 || batch[t] != batch[t - 1]) starts[batch[t]] = t;
}
__global__ void nvalid_kernel(const float* __restrict__ cnt, int* nv, int g) {
    TGRID if (t >= g) return;
    int c = (int)(cnt[t] + 0.5f);
    nv[t] = c < MAXN_T ? c : MAXN_T;
}
__global__ void scatter_dense(const float* __restrict__ xf, const int* __restrict__ batch,
                              const int* __restrict__ starts, float* dense, int n) { // n=N*F
    TGRID if (t >= n) return;
    int i = t / F_DIM, f = t % F_DIM, g = batch[i];
    int pos = i - starts[g];
    if (pos < MAXN_T)
        dense[((size_t)g * MAXN_T + pos) * F_DIM + f] = xf[t];
}
__global__ void vec_lin(const float* __restrict__ S, const float* __restrict__ W,
                        const float* __restrict__ b, float* out, int din, int dout) {
    TGRID if (t >= dout) return;
    float acc = b[t];
    for (int d = 0; d < din; ++d) acc += S[d] * W[(size_t)d * dout + t];
    out[t] = acc;
}
// PMA attention (k=1 seed, 8 heads, up to 128 keys). One block (128 thr) per graph.
__global__ __launch_bounds__(F_DIM)
void pma_attention(const float* __restrict__ qv, const float* __restrict__ K,
                   const float* __restrict__ V, const int* __restrict__ nvalid,
                   float* __restrict__ z0) {
    int g = blockIdx.x, f = threadIdx.x;
    __shared__ float s[HEADS_T * MAXN_T];
    __shared__ float q_s[F_DIM];
    q_s[f] = qv[f];
    __syncthreads();
    int nv = nvalid[g];
    const float* Kg = K + (size_t)g * MAXN_T * F_DIM;
    const float* Vg = V + (size_t)g * MAXN_T * F_DIM;
    // thread f computes scores for key k=f, all heads
    for (int h = 0; h < HEADS_T; ++h) {
        float acc = 0.f;
#pragma unroll
        for (int d = 0; d < F_DIM / HEADS_T; ++d)
            acc += q_s[h * (F_DIM / HEADS_T) + d] * Kg[(size_t)f * F_DIM + h * (F_DIM / HEADS_T) + d];
        s[h * MAXN_T + f] = acc * 0.088388347648318f;   // 1/sqrt(128)
    }
    __syncthreads();
    int nveff = (nv == 0) ? MAXN_T : nv;    // empty graph: uniform over padded rows
    if (f < HEADS_T) {
        int h = f;
        if (nv == 0) {
            for (int k = 0; k < MAXN_T; ++k) s[h * MAXN_T + k] = 1.f / (float)MAXN_T;
        } else {
            float m = -1e30f;
            for (int k = 0; k < nv; ++k) m = fmaxf(m, s[h * MAXN_T + k]);
            float sum = 0.f;
            for (int k = 0; k < nv; ++k) { float e = __expf(s[h * MAXN_T + k] - m); s[h * MAXN_T + k] = e; sum += e; }
            float inv = 1.f / sum;
            for (int k = 0; k < nv; ++k) s[h * MAXN_T + k] *= inv;
        }
    }
    __syncthreads();
    int h = f >> 4;   // F/HEADS = 16
    float acc = 0.f;
    for (int k = 0; k < nveff; ++k) acc += s[h * MAXN_T + k] * Vg[(size_t)k * F_DIM + f];
    z0[(size_t)g * F_DIM + f] = q_s[f] + acc;
}
__global__ void add_relu(const float* __restrict__ a, const float* __restrict__ b,
                         float* o, int n) {
    TGRID if (t >= n) return;
    o[t] = a[t] + fmaxf(b[t], 0.f);
}
__global__ void ew_add(const float* __restrict__ a, const float* __restrict__ b,
                       float* o, int n) {
    TGRID if (t >= n) return;
    o[t] = a[t] + b[t];
}
__global__ void ew_addvec(const float* __restrict__ vec, const float* __restrict__ b,
                          float* o, int n, int C) {
    TGRID if (t >= n) return;
    o[t] = vec[t % C] + b[t];
}
__global__ void vae_out(const float* __restrict__ mu, const float* __restrict__ lv,
                        const float* __restrict__ eps, float* out, int n) {
    TGRID if (t >= n) return;
    float m = mu[t], l = lv[t];
    out[t]         = eps[t] * __expf(0.5f * l) + m;   // latent
    out[n + t]     = m;                                // mu
    out[2 * n + t] = l;                                // logvar
}

// ------------------------------ host side ----------------------------------
static inline void run_gemm(hipStream_t s, const float* X, const float* W, const float* b,
                            float* Y, int M, int din, int dout, int act) {
    const int Kc = din < 128 ? din : 128;
    const size_t shmem = (size_t)Kc * dout * sizeof(float);
    dim3 g((M + 127) / 128);
    if (dout == 128) gemm_v3<8><<<g, 256, shmem, s>>>(X, W, b, Y, M, din, act);
    else             gemm_v3<4><<<g, 256, shmem, s>>>(X, W, b, Y, M, din, act);
}
#define EW(kern, n, ...) kern<<<((n) + 255) / 256, 256, 0, stream>>>(__VA_ARGS__)

extern "C" void kernel_launch(void* const* d_in, const int* in_sizes, int n_in,
                              void* d_out, int out_size, void* d_ws, size_t ws_size,
                              hipStream_t stream) {
    (void)in_sizes; (void)n_in; (void)out_size; (void)ws_size;
    const int N = N_NODES, E = N_EDGES, G = N_GRAPH;
    const size_t NF = (size_t)N * F_DIM;
    const size_t GF = (size_t)G * F_DIM;
    const size_t GMF = (size_t)G * MAXN_T * F_DIM;

    // --- inputs ---
    const int*   x_atom = (const int*)d_in[0];
    const int*   x_chir = (const int*)d_in[1];
    const int*   ei     = (const int*)d_in[2];
    const int*   src    = ei;
    const int*   dst    = ei + E;
    const int*   batch  = (const int*)d_in[3];
    const float* eps    = (const float*)d_in[4];
    // params leaves: top-level insertion order, alphabetical within params
    const float* P[128];
    for (int i = 5; i < 110; ++i) P[i] = (const float*)d_in[i];

    // --- workspace carve-up ---
    float* ws    = (float*)d_ws;
    float* hbuf  = ws;               // NF
    float* htmp  = hbuf + NF;        // NF
    float* t1    = htmp + NF;        // NF
    float* agg   = t1 + NF;          // NF
    float* xs123 = agg + NF;         // 3*NF  (x1,x3,x5 ; later kdense)
    float* xp    = xs123 + 3 * NF;   // 3*NF  (xs_p,xm_p,xl_p ; later vdense)
    float* qb    = xp + 3 * NF;      // NF
    float* kb    = qb + NF;          // NF
    float* vb    = kb + NF;          // NF
    float* cc    = vb + NF;          // 3*NF  (concat ; later dense)
    float* segbuf  = cc + 3 * NF;    // GF
    float* segbuf2 = segbuf + GF;    // GF
    float* cntf  = segbuf2 + GF;     // G
    float* msq   = cntf + G;         // G
    float* degf  = msq + G;          // N (degree -> dis in place)
    float* coef  = degf + N;         // E
    float* bnacc = coef + E;         // 2*F
    float* zA    = bnacc + 2 * F_DIM;
    float* zB    = zA + GF;
    float* zC    = zB + GF;
    float* zD    = zC + GF;
    float* muB   = zD + GF;          // G*H
    float* lvB   = muB + (size_t)G * H_DIM;
    float* qvec  = lvB + (size_t)G * H_DIM;   // F
    float* qvec2 = qvec + F_DIM;              // F
    int*   starts = (int*)(qvec2 + F_DIM);    // G ints
    int*   nvalid = starts + G;               // G ints

    float* x1s = xs123, *x3s = xs123 + NF, *x5s = xs123 + 2 * NF;
    float* xs_p = xp, *xm_p = xp + NF, *xl_p = xp + 2 * NF;

    // --- degree / counts (shared by all layers) ---
    hipMemsetAsync(degf, 0, (size_t)N * 4, stream);
    hipMemsetAsync(cntf, 0, (size_t)G * 4, stream);
    EW(deg_kernel, E, dst, degf, E);
    EW(cnt_kernel, N, batch, cntf, N);
    EW(dis_kernel, N, degf, N);                           // degf now holds rsqrt(deg+1)
    EW(coef_kernel, E, src, dst, degf, coef, E);

    // --- embedding ---
    EW(embed_kernel, N * H_DIM, x_atom, x_chir, P[9], P[10], hbuf, N * H_DIM);

    // --- 5 GCN layers ---
    const float* cur = hbuf; int curd = H_DIM;
    float* dsts[5] = { x1s, hbuf, x3s, hbuf, x5s };
    for (int i = 0; i < 5; ++i) {
        const float* Wg = P[53 + 2 * i];
        const float* bg = P[54 + 2 * i];
        run_gemm(stream, cur, Wg, nullptr, htmp, N, curd, F_DIM, 0);       // h = x@W (WMMA)
        hipMemsetAsync(agg, 0, NF * 4, stream);
        EW(scatter_edges, E * F_DIM, htmp, src, dst, coef, agg, E * F_DIM);
        EW(gcn_finish, (int)NF, agg, htmp, degf, bg, t1, (int)NF);          // + self loop + bias
        if (i < 3) {   // GraphNorm
            const float* gb = P[89 + 3 * i], *gms = P[90 + 3 * i], *gw = P[91 + 3 * i];
            hipMemsetAsync(segbuf, 0, GF * 4, stream);
            hipMemsetAsync(segbuf2, 0, GF * 4, stream);
            EW(seg_sum_feat, (int)NF, t1, batch, segbuf, (int)NF);
            EW(gn_center, (int)NF, t1, batch, segbuf, cntf, gms, htmp, segbuf2, (int)NF);
            EW(gn_final, (int)NF, htmp, batch, segbuf2, cntf, gw, gb, htmp, (int)NF);
        } else {       // BatchNorm (bn4 / bn5)
            const float* bb = (i == 3) ? P[5] : P[7];
            const float* bgm = (i == 3) ? P[6] : P[8];
            hipMemsetAsync(bnacc, 0, 2 * F_DIM * 4, stream);
            bn_stats<<<F_DIM, 256, 0, stream>>>(t1, bnacc, N, F_DIM);
            EW(bn_apply, (int)NF, t1, bnacc, bgm, bb, htmp, N, F_DIM);
        }
        // PairNorm + ReLU
        hipMemsetAsync(segbuf, 0, GF * 4, stream);
        hipMemsetAsync(msq, 0, (size_t)G * 4, stream);
        EW(seg_sum_feat, (int)NF, htmp, batch, segbuf, (int)NF);
        EW(pn_center, (int)NF, htmp, batch, segbuf, cntf, t1, msq, (int)NF);
        EW(pn_final_relu, (int)NF, t1, batch, msq, cntf, dsts[i], (int)NF);
        cur = dsts[i]; curd = F_DIM;
    }

    // --- fusion projections (WMMA) ---
    run_gemm(stream, x1s, P[31], P[32], xs_p, N, F_DIM, F_DIM, 0);   // ps
    run_gemm(stream, x3s, P[29], P[30], xm_p, N, F_DIM, F_DIM, 0);   // pm
    run_gemm(stream, x5s, P[27], P[28], xl_p, N, F_DIM, F_DIM, 0);   // pl

    // cross attention: x-> k,v ; y-> q ; out -> relu(ln(o+y))
    struct CA { const float* x; const float* y; int base; float* out; };
    CA cas[3] = { { xs_p, xm_p, 43, x1s },     // sm
                  { xs_p, xl_p, 33, x3s },     // sl
                  { xm_p, xl_p, 17, x5s } };   // ml
    for (int ci = 0; ci < 3; ++ci) {
        int bidx = cas[ci].base;     // k.W, k.b, ln_b, ln_g, o.W, o.b, q.W, q.b, v.W, v.b
        run_gemm(stream, cas[ci].y, P[bidx + 6], P[bidx + 7], qb, N, F_DIM, F_DIM, 0); // q
        run_gemm(stream, cas[ci].x, P[bidx + 0], P[bidx + 1], kb, N, F_DIM, F_DIM, 0); // k
        run_gemm(stream, cas[ci].x, P[bidx + 8], P[bidx + 9], vb, N, F_DIM, F_DIM, 0); // v
        EW(ew_sigmul, (int)NF, qb, kb, vb, t1, (int)NF);
        run_gemm(stream, t1, P[bidx + 4], P[bidx + 5], kb, N, F_DIM, F_DIM, 0);        // o
        add_ln_act<<<N, F_DIM, 0, stream>>>(kb, cas[ci].y, P[bidx + 3], P[bidx + 2],
                                            cas[ci].out, 1);
    }
    EW(concat3, N * 3 * F_DIM, x1s, x3s, x5s, cc, N * 3 * F_DIM);
    run_gemm(stream, cc, P[13], P[14], kb, N, 3 * F_DIM, F_DIM, 2);   // gate = sigmoid(lin)
    run_gemm(stream, cc, P[11], P[12], vb, N, 3 * F_DIM, F_DIM, 0);   // final
    EW(fuse_gate, (int)NF, kb, vb, xm_p, t1, (int)NF);
    add_ln_act<<<N, F_DIM, 0, stream>>>(t1, xm_p, P[16], P[15], hbuf, 1);  // xf in hbuf

    // --- GMT: build dense [G,128,128] (reuse cc region) ---
    float* dense = cc;
    float* kdense = xs123;
    float* vdense = xp;
    EW(starts_kernel, N, batch, starts, N);
    EW(nvalid_kernel, G, cntf, nvalid, G);
    hipMemsetAsync(dense, 0, GMF * 4, stream);
    EW(scatter_dense, (int)NF, hbuf, batch, starts, dense, (int)NF);

    // PMA1
    const int GM = G * MAXN_T;
    run_gemm(stream, dense, P[64], P[65], kdense, GM, F_DIM, F_DIM, 0);   // k
    run_gemm(stream, dense, P[70], P[71], vdense, GM, F_DIM, F_DIM, 0);   // v
    vec_lin<<<1, F_DIM, 0, stream>>>(P[63], P[68], P[69], qvec, F_DIM, F_DIM);  // q = S@Wq+b
    pma_attention<<<G, F_DIM, 0, stream>>>(qvec, kdense, vdense, nvalid, zA);
    run_gemm(stream, zA, P[66], P[67], zB, G, F_DIM, F_DIM, 0);           // o
    EW(add_relu, (int)GF, zA, zB, zC, (int)GF);                           // z = out + relu(o)

    // SAB (nq=nk=1: attention collapses to q+v)
    run_gemm(stream, zC, P[85], P[86], zA, G, F_DIM, F_DIM, 0);  // q
    run_gemm(stream, zC, P[87], P[88], zB, G, F_DIM, F_DIM, 0);  // v
    EW(ew_add, (int)GF, zA, zB, zA, (int)GF);
    run_gemm(stream, zA, P[83], P[84], zB, G, F_DIM, F_DIM, 0);  // o
    EW(add_relu, (int)GF, zA, zB, zC, (int)GF);

    // PMA2 (single key: out = q_const + v)
    vec_lin<<<1, F_DIM, 0, stream>>>(P[72], P[77], P[78], qvec2, F_DIM, F_DIM);
    run_gemm(stream, zC, P[79], P[80], zB, G, F_DIM, F_DIM, 0);  // v
    EW(ew_addvec, (int)GF, qvec2, zB, zA, (int)GF, F_DIM);
    run_gemm(stream, zA, P[75], P[76], zB, G, F_DIM, F_DIM, 0);  // o
    EW(add_relu, (int)GF, zA, zB, zC, (int)GF);

    // projection head
    run_gemm(stream, zC, P[102], P[103], zA, G, F_DIM, H_DIM, 1);  // relu(l1)
    run_gemm(stream, zA, P[104], P[105], zB, G, H_DIM, H_DIM, 0);  // l2

    // mu / logvar heads with train-mode BatchNorm
    const int GH = G * H_DIM;
    run_gemm(stream, zB, P[100], P[101], zA, G, H_DIM, H_DIM, 0);  // mu.lin
    hipMemsetAsync(bnacc, 0, 2 * H_DIM * 4, stream);
    bn_stats<<<H_DIM, 256, 0, stream>>>(zA, bnacc, G, H_DIM);
    EW(bn_apply, GH, zA, bnacc, P[99], P[98], muB, G, H_DIM);
    run_gemm(stream, zB, P[108], P[109], zD, G, H_DIM, H_DIM, 0);  // var.lin
    hipMemsetAsync(bnacc, 0, 2 * H_DIM * 4, stream);
    bn_stats<<<H_DIM, 256, 0, stream>>>(zD, bnacc, G, H_DIM);
    EW(bn_apply, GH, zD, bnacc, P[107], P[106], lvB, G, H_DIM);

    // latent = eps*exp(0.5*logvar)+mu ; d_out = [latent | mu | logvar]
    EW(vae_out, GH, muB, lvB, eps, (float*)d_out, GH);
}